// GCN_Classification_14817637171423
// MI455X (gfx1250) — compile-verified
//
#include <hip/hip_runtime.h>
#include <hip/hip_bf16.h>

typedef float v2f __attribute__((ext_vector_type(2)));
typedef float v8f __attribute__((ext_vector_type(8)));

// ---------------------------------------------------------------------------
// Degree / normalization
// ---------------------------------------------------------------------------
__global__ void k_deg_init(float* __restrict__ deg, int n) {
    int i = blockIdx.x * blockDim.x + threadIdx.x;
    if (i < n) deg[i] = 1.0f;  // self-loop contributes 1
}

__global__ void k_deg_edges(const int* __restrict__ dst, float* __restrict__ deg, int nE) {
    int e = blockIdx.x * blockDim.x + threadIdx.x;
    if (e < nE) atomicAdd(&deg[dst[e]], 1.0f);
}

__global__ void k_rsqrt(float* __restrict__ deg, int n) {
    int i = blockIdx.x * blockDim.x + threadIdx.x;
    if (i < n) deg[i] = rsqrtf(deg[i]);  // deg >= 1 always
}

// ---------------------------------------------------------------------------
// WMMA fp32 GEMM: C[M,N] = A[M,K] * B[K,N]   (row-major, K,N compile-time)
// Block = 256 threads = 8 waves; each wave owns a 16x16 tile of a 32x64 block.
// ---------------------------------------------------------------------------
template <int K, int N>
__launch_bounds__(256)
__global__ void gemm_wmma(const float* __restrict__ A, const float* __restrict__ B,
                          float* __restrict__ C, int M) {
    constexpr int BM = 32, BN = 64, BK = 64;
    constexpr int AS = BK + 4;  // 68: conflict-free strided A reads
    constexpr int BS = BN + 8;  // 72: lanes 0-15 / 16-31 hit disjoint bank ranges
    __shared__ float As[BM * AS];
    __shared__ float Bs[BK * BS];

    const int tid  = threadIdx.x;
    const int lane = tid & 31;
    const int wave = tid >> 5;
    const int block_m = blockIdx.x * BM;
    const int block_n = blockIdx.y * BN;
    const int tile_m = (wave >> 2) * 16;  // 0 / 16
    const int tile_n = (wave & 3) * 16;   // 0..48

    v8f acc = {};

    const int arow  = tile_m + (lane & 15);
    const int bcol  = tile_n + (lane & 15);
    const int khalf = (lane >> 4) * 2;  // lanes 16-31 hold K+2,K+3

    for (int k0 = 0; k0 < K; k0 += BK) {
        // stage A tile (BM x BK), clamped rows at the M edge
        for (int i = tid; i < BM * BK / 4; i += 256) {
            int r = (i * 4) / BK, c = (i * 4) % BK;
            int gr = block_m + r; if (gr >= M) gr = M - 1;
            float4 v = *(const float4*)(A + (size_t)gr * K + k0 + c);
            *(float4*)(&As[r * AS + c]) = v;
        }
        // stage B tile (BK x BN)
        for (int i = tid; i < BK * BN / 4; i += 256) {
            int r = (i * 4) / BN, c = (i * 4) % BN;
            float4 v = *(const float4*)(B + (size_t)(k0 + r) * N + block_n + c);
            *(float4*)(&Bs[r * BS + c]) = v;
        }
        __syncthreads();

#pragma unroll
        for (int kk = 0; kk < BK; kk += 4) {
            v2f a, b;
            a.x = As[arow * AS + kk + khalf];
            a.y = As[arow * AS + kk + khalf + 1];
            b.x = Bs[(kk + khalf) * BS + bcol];
            b.y = Bs[(kk + khalf + 1) * BS + bcol];
            acc = __builtin_amdgcn_wmma_f32_16x16x4_f32(
                false, a, false, b, (short)0, acc, false, false);
        }
        __syncthreads();
    }

    // D layout: VGPR v -> row (lane>=16 ? 8 : 0)+v, col = lane&15
    const int crow = block_m + tile_m + (lane >> 4) * 8;
    const int ccol = block_n + tile_n + (lane & 15);
#pragma unroll
    for (int v = 0; v < 8; v++) {
        int gr = crow + v;
        if (gr < M) C[(size_t)gr * N + ccol] = acc[v];
    }
}

// ---------------------------------------------------------------------------
// Aggregation: agg[n] = dinv[n]^2 * h[n]  +  sum_{e: dst=n} dinv[s]dinv[n] h[s]
// ---------------------------------------------------------------------------
template <int F>
__global__ void k_agg_init(const float* __restrict__ h, const float* __restrict__ dinv,
                           float* __restrict__ agg) {
    int n = blockIdx.x;
    float w = dinv[n]; w *= w;
    int f = threadIdx.x;
    agg[(size_t)n * F + f] = h[(size_t)n * F + f] * w;
}

template <int F>
__global__ void k_agg_edges(const float* __restrict__ h, const int* __restrict__ src,
                            const int* __restrict__ dst, const float* __restrict__ dinv,
                            float* __restrict__ agg) {
    int e = blockIdx.x;
    int s = src[e], d = dst[e];
    float w = dinv[s] * dinv[d];
    int f = threadIdx.x;
    atomicAdd(&agg[(size_t)d * F + f], h[(size_t)s * F + f] * w);
}

template <int F>
__global__ void k_bias_relu(float* __restrict__ h, const float* __restrict__ b) {
    int n = blockIdx.x, f = threadIdx.x;
    size_t i = (size_t)n * F + f;
    h[i] = fmaxf(h[i] + b[f], 0.0f);
}

// ---------------------------------------------------------------------------
// Pooling + classifier head
// ---------------------------------------------------------------------------
__global__ void k_pool_init(float* __restrict__ p, int n) {
    int i = blockIdx.x * blockDim.x + threadIdx.x;
    if (i < n) p[i] = 0.0f;
}

__global__ void k_pool_accum(const float* __restrict__ h, const int* __restrict__ batch,
                             float* __restrict__ pool, float* __restrict__ cnt) {
    int n = blockIdx.x;
    int g = batch[n];
    int f = threadIdx.x;  // 128
    atomicAdd(&pool[g * 128 + f], h[(size_t)n * 128 + f]);
    if (f == 0) atomicAdd(&cnt[g], 1.0f);
}

__global__ void k_final(const float* __restrict__ pool, const float* __restrict__ cnt,
                        const float* __restrict__ Wl, const float* __restrict__ bl,
                        float* __restrict__ out) {
    int t = threadIdx.x;  // 128 = 64 graphs * 2 classes
    int g = t >> 1, c = t & 1;
    float inv = 1.0f / fmaxf(cnt[g], 1.0f);
    float s = bl[c];
    for (int f = 0; f < 128; f++)
        s += pool[g * 128 + f] * inv * Wl[c * 128 + f];
    out[t] = s;
}

// ---------------------------------------------------------------------------
extern "C" void kernel_launch(void* const* d_in, const int* in_sizes, int n_in,
                              void* d_out, int out_size, void* d_ws, size_t ws_size,
                              hipStream_t stream) {
    const float* x    = (const float*)d_in[0];
    const int*   ei   = (const int*)d_in[1];
    const int*   batch= (const int*)d_in[2];
    const float* W1   = (const float*)d_in[3];
    const float* b1   = (const float*)d_in[4];
    const float* W2   = (const float*)d_in[5];
    const float* b2   = (const float*)d_in[6];
    const float* Wl   = (const float*)d_in[7];
    const float* bl   = (const float*)d_in[8];

    const int D_IN = 128, H1 = 256, H2 = 128;
    int nN = in_sizes[0] / D_IN;       // 50000
    int nE = in_sizes[1] / 2;          // 800000
    const int* src = ei;
    const int* dst = ei + nE;

    char* ws = (char*)d_ws;
    float* dinv = (float*)ws;                                   // nN floats
    size_t offA = (((size_t)nN * 4) + 1023) & ~(size_t)1023;
    float* bufA = (float*)(ws + offA);                          // nN * 256
    size_t offB = offA + (size_t)nN * 256 * 4;
    float* bufB = (float*)(ws + offB);                          // nN * 256
    size_t offP = offB + (size_t)nN * 256 * 4;
    float* pool = (float*)(ws + offP);                          // 64*128
    float* cnt  = pool + 64 * 128;                              // 64

    // normalization
    k_deg_init <<<(nN + 255) / 256, 256, 0, stream>>>(dinv, nN);
    k_deg_edges<<<(nE + 255) / 256, 256, 0, stream>>>(dst, dinv, nE);
    k_rsqrt    <<<(nN + 255) / 256, 256, 0, stream>>>(dinv, nN);

    // layer 1: h1 = relu(aggregate(x @ W1) + b1)
    {
        dim3 grid((nN + 31) / 32, H1 / 64);
        gemm_wmma<128, 256><<<grid, 256, 0, stream>>>(x, W1, bufA, nN);
    }
    k_agg_init<256> <<<nN, 256, 0, stream>>>(bufA, dinv, bufB);
    k_agg_edges<256><<<nE, 256, 0, stream>>>(bufA, src, dst, dinv, bufB);
    k_bias_relu<256><<<nN, 256, 0, stream>>>(bufB, b1);

    // layer 2: h2 = relu(aggregate(h1 @ W2) + b2)
    {
        dim3 grid((nN + 31) / 32, H2 / 64);
        gemm_wmma<256, 128><<<grid, 256, 0, stream>>>(bufB, W2, bufA, nN);
    }
    k_agg_init<128> <<<nN, 128, 0, stream>>>(bufA, dinv, bufB);
    k_agg_edges<128><<<nE, 128, 0, stream>>>(bufA, src, dst, dinv, bufB);
    k_bias_relu<128><<<nN, 128, 0, stream>>>(bufB, b2);

    // mean pool + head
    k_pool_init <<<(64 * 128 + 64 + 255) / 256, 256, 0, stream>>>(pool, 64 * 128 + 64);
    k_pool_accum<<<nN, 128, 0, stream>>>(bufB, batch, pool, cnt);
    k_final     <<<1, 128, 0, stream>>>(pool, cnt, Wl, bl, (float*)d_out);
}